// InnerProduct_6227702579222
// MI455X (gfx1250) — compile-verified
//
#include <hip/hip_runtime.h>
#include <hip/hip_bf16.h>

typedef __attribute__((ext_vector_type(2))) float v2f;
typedef __attribute__((ext_vector_type(8))) float v8f;

#define BATCH   16384
#define NFIELDS 32
#define EMBED   64
#define NPAIR   496   // 32*31/2

// Flat index of (i,j), i<j, in row-major strict-upper-triangle order
// (matches jnp.triu_indices(32, k=1)).
__device__ __forceinline__ int tri_idx(int i, int j) {
    return i * 31 - (i * (i - 1)) / 2 + (j - i - 1);
}

__global__ __launch_bounds__(256) void gram_triu_wmma(const float* __restrict__ x,
                                                      float* __restrict__ out) {
    // One batch per wave32. Grid is sized exactly: no divergence, EXEC all-ones.
    const int wave = blockIdx.x * (blockDim.x >> 5) + (threadIdx.x >> 5);
    const int lane = threadIdx.x & 31;
    const int b    = wave;

    const float* xb = x + (size_t)b * (NFIELDS * EMBED);

    // CDNA5 V_WMMA_F32_16X16X4_F32 operand layout (identical for A and B here):
    // lane L holds element [row = L&15][col = 4k + 2*(L>>4) + {v0,v1}].
    const int h = lane >> 4;    // 0: K pair {0,1}, 1: K pair {2,3}
    const int r = lane & 15;    // row within 16-row tile

    const float* p0 = xb + r * EMBED + 2 * h;           // rows 0..15
    const float* p1 = xb + (16 + r) * EMBED + 2 * h;    // rows 16..31

    v2f t0[16], t1[16];
#pragma unroll
    for (int k = 0; k < 16; ++k) {
        t0[k] = *(const v2f*)(p0 + 4 * k);   // global_load_b64
        t1[k] = *(const v2f*)(p1 + 4 * k);
    }

    v8f acc00 = {};   // gram tile rows 0-15,  cols 0-15
    v8f acc01 = {};   // gram tile rows 0-15,  cols 16-31
    v8f acc11 = {};   // gram tile rows 16-31, cols 16-31

    // D = A*B + C ; same register data feeds A (row tile) and B (col tile,
    // transpose implicit in layout). Tile (1,0) skipped: gram is symmetric.
#pragma unroll
    for (int k = 0; k < 16; ++k) {
        acc00 = __builtin_amdgcn_wmma_f32_16x16x4_f32(
            false, t0[k], false, t0[k], (short)0, acc00, false, false);
        acc01 = __builtin_amdgcn_wmma_f32_16x16x4_f32(
            false, t0[k], false, t1[k], (short)0, acc01, false, false);
        acc11 = __builtin_amdgcn_wmma_f32_16x16x4_f32(
            false, t1[k], false, t1[k], (short)0, acc11, false, false);
    }

    // C/D layout: lane 0-15 -> (M=v,   N=lane), lane 16-31 -> (M=v+8, N=lane-16)
    float* ob = out + (size_t)b * NPAIR;
#pragma unroll
    for (int v = 0; v < 8; ++v) {
        const int m = v + 8 * h;   // local row
        const int n = r;           // local col

        if (n > m) ob[tri_idx(m, n)] = acc00[v];              // diag block (0,0)
        ob[tri_idx(m, 16 + n)] = acc01[v];                    // off-diag: always i<j
        if (n > m) ob[tri_idx(16 + m, 16 + n)] = acc11[v];    // diag block (1,1)
    }
}

extern "C" void kernel_launch(void* const* d_in, const int* in_sizes, int n_in,
                              void* d_out, int out_size, void* d_ws, size_t ws_size,
                              hipStream_t stream) {
    const float* x   = (const float*)d_in[0];
    float*       out = (float*)d_out;
    (void)in_sizes; (void)n_in; (void)out_size; (void)d_ws; (void)ws_size;

    // 256 threads = 8 wave32 per block, 1 batch per wave -> 2048 blocks.
    const int waves_per_block = 256 / 32;
    const int blocks = BATCH / waves_per_block;   // 2048, exact
    gram_triu_wmma<<<blocks, 256, 0, stream>>>(x, out);
}